// MDR_local_global_72447508349152
// MI455X (gfx1250) — compile-verified
//
#include <hip/hip_runtime.h>

// ---------------------------------------------------------------------------
// MoE routed projection + LayerNorm + cross-domain logits for MI455X (gfx1250)
//
// Key insight: routing masks are disjoint -> only ONE expert per token.
// Bucket tokens by expert, run a grouped fp32 GEMM with V_WMMA_F32_16X16X4_F32
// (A = gathered token rows, B = W_e^T slice staged in LDS), fuse bias +
// LayerNorm in the epilogue, then a memory-bound gather/dot kernel for logits.
// ---------------------------------------------------------------------------

#define NDOM 4
#define BB   128
#define LL   200
#define DDIM 512
#define NTOK (NDOM * BB * LL)          // 102400 tokens
#define MT   32                        // tokens per workgroup (GEMM M tile)
#define KC   32                        // K chunk staged in LDS
#define TILES_PER_E ((NTOK + MT - 1) / MT)  // 3200
#define SA_STRIDE 36                   // LDS row strides (bank-conflict-free)
#define SB_STRIDE 36
#define SC_STRIDE 520

typedef float v2f __attribute__((ext_vector_type(2)));
typedef float v8f __attribute__((ext_vector_type(8)));

__global__ void zero_cnt_kernel(int* __restrict__ cnt) {
  if (threadIdx.x < NDOM) cnt[threadIdx.x] = 0;
}

// Bucket tokens by routed expert. Slot order is nondeterministic but every
// token's output value is independent of its slot -> d_out is deterministic.
__global__ void route_kernel(const int* __restrict__ lg_dom,
                             int* __restrict__ cnt,
                             int* __restrict__ idxList) {
  int t = blockIdx.x * blockDim.x + threadIdx.x;
  if (t >= NTOK) return;
  int i   = t / (BB * LL);
  int rem = t % (BB * LL);
  int b   = rem / LL;
  int l   = rem % LL;
  int lg  = lg_dom[(b * NDOM + i) * LL + l];   // lg_dom shape [B, ND, L]
  if (lg > 0) {
    int e    = lg - 1;
    int slot = atomicAdd(&cnt[e], 1);
    idxList[e * NTOK + slot] = t;
  }
}

// Grouped GEMM (gathered A rows) + bias + LayerNorm, fp32 WMMA.
// Block = 256 threads = 8 waves. Wave w owns rows (w&1)*16..+15 and a
// 128-column strip (w>>1)*128 as 8 x v_wmma_f32_16x16x4_f32 accumulators.
__global__ __launch_bounds__(256, 1) void moe_gemm_ln_kernel(
    const float* __restrict__ X,      // log_feats  [NTOK, D]
    const float* __restrict__ W,      // W_maps     [ND, D, D]  (row n, col d)
    const float* __restrict__ bias,   // b_maps     [ND, D]
    const float* __restrict__ gamma,  // [D]
    const float* __restrict__ beta,   // [D]
    const int*   __restrict__ cnt,    // [ND]
    const int*   __restrict__ idxList,// [ND, NTOK]
    float* __restrict__ mapped)       // [NTOK, D]
{
  __shared__ float sA[MT * SA_STRIDE];     //  4.6 KB
  __shared__ float sB[DDIM * SB_STRIDE];   // 73.7 KB, reused as C buffer
  float* sC = sB;                          // C: [MT][SC_STRIDE] fits in sB

  const int e    = blockIdx.x / TILES_PER_E;
  const int tile = blockIdx.x % TILES_PER_E;
  const int count = cnt[e];
  const int row0  = tile * MT;
  if (row0 >= count) return;                       // block-uniform early exit
  const int nvalid = min(MT, count - row0);
  const int* myIdx = idxList + e * NTOK + row0;

  const int tid   = threadIdx.x;
  const int w     = tid >> 5;
  const int lane  = tid & 31;
  const int lh    = lane >> 4;                     // lane half (0/1)
  const int l15   = lane & 15;
  const int rhalf = (w & 1) * 16;                  // this wave's 16 M-rows
  const int cg    = (w >> 1) * 128;                // this wave's 128 N-cols

  v8f acc[8] = {};

  const float* We = W + (size_t)e * DDIM * DDIM;

  for (int kc = 0; kc < DDIM; kc += KC) {
    // --- stage A chunk: 32 rows x 32 K (gathered; invalid rows -> 0) -------
    {
      int m  = tid >> 3;                 // 0..31
      int kq = (tid & 7) << 2;           // 0,4,..,28
      float4 v = make_float4(0.f, 0.f, 0.f, 0.f);
      if (m < nvalid) {
        int t = myIdx[m];
        v = *(const float4*)(X + (size_t)t * DDIM + kc + kq);
      }
      *(float4*)(&sA[m * SA_STRIDE + kq]) = v;
    }
    // --- stage B chunk: all 512 output rows x 32 K of W_e (L2-resident) ----
    #pragma unroll
    for (int r = 0; r < 16; ++r) {
      int fi = r * 256 + tid;            // 4096 float4s total
      int n  = fi >> 3;
      int kq = (fi & 7) << 2;
      float4 v = *(const float4*)(We + (size_t)n * DDIM + kc + kq);
      *(float4*)(&sB[n * SB_STRIDE + kq]) = v;
    }
    __syncthreads();

    // --- 8 k-steps of V_WMMA_F32_16X16X4_F32 -------------------------------
    #pragma unroll
    for (int kk = 0; kk < KC; kk += 4) {
      int kl = kk + (lh << 1);           // A/B frag: lanes 16-31 hold K+2,K+3
      v2f a = *(const v2f*)(&sA[(rhalf + l15) * SA_STRIDE + kl]);
      #pragma unroll
      for (int tt = 0; tt < 8; ++tt) {
        v2f bfr = *(const v2f*)(&sB[(cg + tt * 16 + l15) * SB_STRIDE + kl]);
        acc[tt] = __builtin_amdgcn_wmma_f32_16x16x4_f32(
            false, a, false, bfr, (short)0, acc[tt], false, false);
      }
    }
    __syncthreads();
  }

  // --- epilogue: C + bias into LDS (C/D frag: vgpr v -> row v + 8*lanehalf) -
  #pragma unroll
  for (int tt = 0; tt < 8; ++tt) {
    int n = cg + tt * 16 + l15;
    float bv = bias[e * DDIM + n];
    #pragma unroll
    for (int v = 0; v < 8; ++v) {
      int m = rhalf + v + (lh << 3);
      sC[m * SC_STRIDE + n] = acc[tt][v] + bv;
    }
  }
  __syncthreads();

  // --- LayerNorm: wave w handles rows w*4 .. w*4+3 (two-pass, eps=1e-8) ----
  for (int rr = 0; rr < 4; ++rr) {
    int m = w * 4 + rr;
    if (m >= nvalid) continue;           // wave-uniform
    float vals[16];
    float s = 0.f;
    #pragma unroll
    for (int q = 0; q < 16; ++q) {
      float x = sC[m * SC_STRIDE + lane + 32 * q];
      vals[q] = x;
      s += x;
    }
    #pragma unroll
    for (int off = 16; off >= 1; off >>= 1) s += __shfl_xor(s, off, 32);
    float mu = s * (1.0f / DDIM);
    float s2 = 0.f;
    #pragma unroll
    for (int q = 0; q < 16; ++q) { float d = vals[q] - mu; s2 += d * d; }
    #pragma unroll
    for (int off = 16; off >= 1; off >>= 1) s2 += __shfl_xor(s2, off, 32);
    float inv = rsqrtf(s2 * (1.0f / DDIM) + 1e-8f);
    int t = myIdx[m];
    float* out = mapped + (size_t)t * DDIM;
    #pragma unroll
    for (int q = 0; q < 16; ++q) {
      int col = lane + 32 * q;
      out[col] = (vals[q] - mu) * inv * gamma[col] + beta[col];
    }
  }
}

// One wave per token: pad rows -> zero-fill mapped + zero logits;
// routed rows -> gather pos/neg embedding rows and dot with mapped.
__global__ __launch_bounds__(256) void logits_kernel(
    float* __restrict__ mapped,
    const float* __restrict__ emb,
    const int* __restrict__ lg_dom,
    const int* __restrict__ pos_idx,
    const int* __restrict__ neg_idx,
    float* __restrict__ posOut,
    float* __restrict__ negOut)
{
  int w    = threadIdx.x >> 5;
  int lane = threadIdx.x & 31;
  int t    = blockIdx.x * 8 + w;
  if (t >= NTOK) return;
  int i   = t / (BB * LL);
  int rem = t % (BB * LL);
  int b   = rem / LL;
  int l   = rem % LL;
  int li  = (b * NDOM + i) * LL + l;     // [B, ND, L] index
  int lg  = lg_dom[li];
  float* mrow = mapped + (size_t)t * DDIM;
  if (lg == 0) {
    float4 z = make_float4(0.f, 0.f, 0.f, 0.f);
    #pragma unroll
    for (int q = 0; q < 4; ++q)
      *(float4*)(mrow + (q * 32 + lane) * 4) = z;
    if (lane == 0) { posOut[t] = 0.f; negOut[t] = 0.f; }
    return;
  }
  const float* prow = emb + (size_t)pos_idx[li] * DDIM;
  const float* nrow = emb + (size_t)neg_idx[li] * DDIM;
  float dp = 0.f, dn = 0.f;
  #pragma unroll
  for (int q = 0; q < 4; ++q) {
    float4 mv = *(const float4*)(mrow + (q * 32 + lane) * 4);
    float4 pv = *(const float4*)(prow + (q * 32 + lane) * 4);
    float4 nv = *(const float4*)(nrow + (q * 32 + lane) * 4);
    dp += mv.x * pv.x + mv.y * pv.y + mv.z * pv.z + mv.w * pv.w;
    dn += mv.x * nv.x + mv.y * nv.y + mv.z * nv.z + mv.w * nv.w;
  }
  #pragma unroll
  for (int off = 16; off >= 1; off >>= 1) {
    dp += __shfl_xor(dp, off, 32);
    dn += __shfl_xor(dn, off, 32);
  }
  if (lane == 0) { posOut[t] = dp; negOut[t] = dn; }
}

extern "C" void kernel_launch(void* const* d_in, const int* in_sizes, int n_in,
                              void* d_out, int out_size, void* d_ws, size_t ws_size,
                              hipStream_t stream) {
  const float* log_feats = (const float*)d_in[0];
  const float* W_maps    = (const float*)d_in[1];
  const float* b_maps    = (const float*)d_in[2];
  const float* emb       = (const float*)d_in[3];
  const float* ln_gamma  = (const float*)d_in[4];
  const float* ln_beta   = (const float*)d_in[5];
  const int*   lg_dom    = (const int*)d_in[6];
  const int*   pos_idx   = (const int*)d_in[7];
  const int*   neg_idx   = (const int*)d_in[8];

  float* mapped = (float*)d_out;                     // [ND*B*L, D]
  float* posOut = mapped + (size_t)NTOK * DDIM;      // [ND*B*L]
  float* negOut = posOut + NTOK;                     // [ND*B*L]

  int* cnt     = (int*)d_ws;                         // 4 counters (64B pad)
  int* idxList = cnt + 16;                           // [ND, NTOK] ints

  zero_cnt_kernel<<<1, 32, 0, stream>>>(cnt);
  route_kernel<<<(NTOK + 255) / 256, 256, 0, stream>>>(lg_dom, cnt, idxList);
  moe_gemm_ln_kernel<<<NDOM * TILES_PER_E, 256, 0, stream>>>(
      log_feats, W_maps, b_maps, ln_gamma, ln_beta, cnt, idxList, mapped);
  logits_kernel<<<NTOK / 8, 256, 0, stream>>>(
      mapped, emb, lg_dom, pos_idx, neg_idx, posOut, negOut);
}